// LengthRegulator_609885356137
// MI455X (gfx1250) — compile-verified
//
#include <hip/hip_runtime.h>
#include <hip/hip_bf16.h>
#include <stdint.h>

// LengthRegulator: duration-based repeat-expand.
//   hiddens:  [B=32, T=1024, D=512] float32
//   durations:[B, T] int32 in [0, 8)
//   out:      [B, max_len, D] float32, zero-padded past sum(dur) per batch.
// Pure gather / streaming-store problem: ~236 MB stores, ~64 MB unique reads
// (fits in 192 MB L2). Bandwidth bound -> use CDNA5 async global<->LDS DMA
// (global_load_async_to_lds_b128 / global_store_async_from_lds_b128) for the
// row copies, tracked with ASYNCcnt.

#define LR_B 32
#define LR_T 1024
#define LR_D 512
#define ROW_BYTES (LR_D * 4)        // 2048 bytes per row
#define WAVES_PER_BLOCK 8

// ---------------------------------------------------------------------------
// Phase 1: per-batch inclusive cumsum of durations. One wave32 per batch.
// Lane L owns durations[b, L*32 .. L*32+31]; wave-scan of chunk sums via
// __shfl_up, then each lane emits its 32 inclusive prefix values.
// ---------------------------------------------------------------------------
__global__ __launch_bounds__(32)
void lr_scan(const int* __restrict__ dur, int* __restrict__ csum) {
    const int b    = blockIdx.x;
    const int lane = threadIdx.x;          // 0..31 (wave32)
    const int* drow = dur + b * LR_T;

    int v[32];
    int chunk = 0;
#pragma unroll
    for (int k = 0; k < 32; ++k) {
        v[k] = drow[lane * 32 + k];
        chunk += v[k];
    }

    // inclusive wave scan of chunk sums
    int s = chunk;
#pragma unroll
    for (int d = 1; d < 32; d <<= 1) {
        int n = __shfl_up(s, d, 32);
        if (lane >= d) s += n;
    }

    int run = s - chunk;                   // exclusive prefix at chunk start
    int* crow = csum + b * LR_T;
#pragma unroll
    for (int k = 0; k < 32; ++k) {
        run += v[k];
        crow[lane * 32 + k] = run;         // inclusive cumsum
    }
}

// ---------------------------------------------------------------------------
// Phase 2: one wave per output row. Binary search csum for the owning token
// (searchsorted side='right'), then DMA the 2 KB row via async LDS path.
// Masked (padding) rows store zeros directly.
// ---------------------------------------------------------------------------
__global__ __launch_bounds__(WAVES_PER_BLOCK * 32)
void lr_copy(const float* __restrict__ hid, const int* __restrict__ csum,
             float* __restrict__ out, int max_len, int nrows) {
    __shared__ char buf[WAVES_PER_BLOCK][ROW_BYTES];

    const int lane = threadIdx.x & 31;
    const int wave = threadIdx.x >> 5;
    const int row  = blockIdx.x * WAVES_PER_BLOCK + wave;   // wave-uniform
    if (row >= nrows) return;

    const int b = row / max_len;
    const int j = row - b * max_len;
    const int* crow  = csum + b * LR_T;
    const int  total = crow[LR_T - 1];

    float* dst = out + (size_t)row * LR_D;

    if (j >= total) {
        // zero-pad: 4 x b128 per lane covers 2048 bytes wave-wide
        float4 z = make_float4(0.f, 0.f, 0.f, 0.f);
        float4* d4 = (float4*)dst;
#pragma unroll
        for (int k = 0; k < 4; ++k) d4[lane + k * 32] = z;
        return;
    }

    // searchsorted(crow, j, side='right'): first index with crow[idx] > j.
    // All lanes search the same (L2-cached) row; 10 iterations.
    int lo = 0, hi = LR_T;
    while (lo < hi) {
        int mid = (lo + hi) >> 1;
        if (crow[mid] <= j) lo = mid + 1; else hi = mid;
    }
    const int tok = (lo < LR_T) ? lo : (LR_T - 1);

    const char* src  = (const char*)(hid + ((size_t)b * LR_T + tok) * LR_D) + lane * 16;
    char*       dstc = (char*)dst + lane * 16;
    unsigned lds_off = (unsigned)(uintptr_t)(void*)&buf[wave][lane * 16];

    // Async DMA: global -> LDS. Instruction offset applies to BOTH the LDS and
    // global addresses (CDNA5 ISA), so one base covers the row.
    asm volatile(
        "global_load_async_to_lds_b128 %0, %1, off\n\t"
        "global_load_async_to_lds_b128 %0, %1, off offset:512\n\t"
        "global_load_async_to_lds_b128 %0, %1, off offset:1024\n\t"
        "global_load_async_to_lds_b128 %0, %1, off offset:1536\n\t"
        "s_wait_asynccnt 0x0"
        :: "v"(lds_off), "v"(src) : "memory");

    // Async DMA: LDS -> global (no VGPR staging).
    asm volatile(
        "global_store_async_from_lds_b128 %0, %1, off\n\t"
        "global_store_async_from_lds_b128 %0, %1, off offset:512\n\t"
        "global_store_async_from_lds_b128 %0, %1, off offset:1024\n\t"
        "global_store_async_from_lds_b128 %0, %1, off offset:1536\n\t"
        "s_wait_asynccnt 0x0"
        :: "v"(dstc), "v"(lds_off) : "memory");
}

// ---------------------------------------------------------------------------
extern "C" void kernel_launch(void* const* d_in, const int* in_sizes, int n_in,
                              void* d_out, int out_size, void* d_ws, size_t ws_size,
                              hipStream_t stream) {
    const float* hiddens   = (const float*)d_in[0];
    const int*   durations = (const int*)d_in[1];
    float*       out       = (float*)d_out;

    int* csum = (int*)d_ws;                      // B*T ints = 128 KB scratch

    const int max_len = out_size / (LR_B * LR_D);
    const int nrows   = LR_B * max_len;

    lr_scan<<<LR_B, 32, 0, stream>>>(durations, csum);

    const int blocks = (nrows + WAVES_PER_BLOCK - 1) / WAVES_PER_BLOCK;
    lr_copy<<<blocks, WAVES_PER_BLOCK * 32, 0, stream>>>(hiddens, csum, out,
                                                         max_len, nrows);
}